// GNNModel_26809185862017
// MI455X (gfx1250) — compile-verified
//
#include <hip/hip_runtime.h>

#define NUM_TEAMS 100000
#define EMBED     32
#define NEDGES    2000000
#define BATCH     524288
#define NCONV     3

typedef __attribute__((ext_vector_type(2))) float v2f;
typedef __attribute__((ext_vector_type(8))) float v8f;

__device__ __forceinline__ float lrelu(float x) {
    // leaky_relu slope 0.01: for x<0, 0.01x > x, so max() implements it exactly
    return fmaxf(x, 0.01f * x);
}

// ---------------------------------------------------------------------------
// Zero the aggregation buffer (float4 vectorized)
// ---------------------------------------------------------------------------
__global__ void zero_f4(float4* __restrict__ p, int n4) {
    int i = blockIdx.x * blockDim.x + threadIdx.x;
    if (i < n4) p[i] = make_float4(0.f, 0.f, 0.f, 0.f);
}

// ---------------------------------------------------------------------------
// Edge gather+scatter: one wave32 per edge, one lane per feature (EMBED==32).
// msg = x[src] * ew  ->  atomic add into agg[dst].
// Table is 12.8MB -> L2-resident; atomics are native global_atomic_add_f32.
// ---------------------------------------------------------------------------
__global__ void edge_scatter(const float* __restrict__ x,
                             const int*   __restrict__ src,
                             const int*   __restrict__ dst,
                             const float* __restrict__ ew,
                             float* __restrict__ agg,
                             int nEdges) {
    long long tid = (long long)blockIdx.x * blockDim.x + threadIdx.x;
    int e    = (int)(tid >> 5);
    int lane = (int)(tid & 31);
    if (e >= nEdges) return;

    // stream-prefetch the edge arrays ahead (emits global_prefetch_b8)
    if (lane == 0 && e + 2048 < nEdges) {
        __builtin_prefetch(&ew[e + 2048], 0, 0);
        __builtin_prefetch(&src[e + 2048], 0, 0);
        __builtin_prefetch(&dst[e + 2048], 0, 0);
    }

    int   s = src[e];          // wave-uniform (e is wave-uniform)
    int   d = dst[e];
    float w = ew[e];
    float v = x[(size_t)s * EMBED + lane] * w;
    unsafeAtomicAdd(&agg[(size_t)d * EMBED + lane], v);
}

// ---------------------------------------------------------------------------
// Fused node transform:
//   x_out = lrelu( agg @ w_rel^T + b_rel + x_in @ w_root^T )
// One wave computes a 16-row x 32-col tile with V_WMMA_F32_16X16X4_F32:
//   2 N-tiles x 8 K-steps x 2 weight matrices = 32 WMMAs / tile.
// f32 VGPR layouts per ISA 7.12.2:
//   A 16x4 : lane m=l&15, half=l>>4; a = {A[m][k0+2h], A[m][k0+2h+1]}
//   B 4x16 : lane n=l&15, half=l>>4; b = {B[k0+2h][n], B[k0+2h+1][n]}
//            with B[k][n] = W[n][k]  (row-major weight, transposed use)
//   D 16x16: d[r] holds (M = r + 8*half, N = n)
// ---------------------------------------------------------------------------
__global__ void conv_transform(const float* __restrict__ xin,
                               const float* __restrict__ agg,
                               const float* __restrict__ w_rel,
                               const float* __restrict__ b_rel,
                               const float* __restrict__ w_root,
                               float* __restrict__ xout,
                               int nRows) {
    const int lane = threadIdx.x & 31;
    const int wave = threadIdx.x >> 5;
    const int tile = blockIdx.x * (blockDim.x >> 5) + wave;
    const int base = tile * 16;
    if (base >= nRows) return;               // whole-wave exit; EXEC stays all-1s

    const int m    = lane & 15;
    const int half = lane >> 4;

    v8f d0 = {0.f, 0.f, 0.f, 0.f, 0.f, 0.f, 0.f, 0.f};   // N = 0..15
    v8f d1 = {0.f, 0.f, 0.f, 0.f, 0.f, 0.f, 0.f, 0.f};   // N = 16..31

    int rowA = base + m;
    if (rowA >= nRows) rowA = nRows - 1;     // clamp ragged tail (store masked below)
    const float* aggRow = agg + (size_t)rowA * EMBED;
    const float* xRow   = xin + (size_t)rowA * EMBED;
    const float* wr0 = w_rel  + (size_t)(m)      * EMBED;  // output col n     (tile 0)
    const float* wr1 = w_rel  + (size_t)(m + 16) * EMBED;  // output col n+16  (tile 1)
    const float* wt0 = w_root + (size_t)(m)      * EMBED;
    const float* wt1 = w_root + (size_t)(m + 16) * EMBED;

    const int ho = 2 * half;

#pragma unroll
    for (int k0 = 0; k0 < EMBED; k0 += 4) {
        v2f a  = *(const v2f*)(aggRow + k0 + ho);
        v2f b0 = *(const v2f*)(wr0    + k0 + ho);
        v2f b1 = *(const v2f*)(wr1    + k0 + ho);
        d0 = __builtin_amdgcn_wmma_f32_16x16x4_f32(false, a, false, b0, (short)0, d0, false, false);
        d1 = __builtin_amdgcn_wmma_f32_16x16x4_f32(false, a, false, b1, (short)0, d1, false, false);
    }
#pragma unroll
    for (int k0 = 0; k0 < EMBED; k0 += 4) {
        v2f a  = *(const v2f*)(xRow + k0 + ho);
        v2f b0 = *(const v2f*)(wt0  + k0 + ho);
        v2f b1 = *(const v2f*)(wt1  + k0 + ho);
        d0 = __builtin_amdgcn_wmma_f32_16x16x4_f32(false, a, false, b0, (short)0, d0, false, false);
        d1 = __builtin_amdgcn_wmma_f32_16x16x4_f32(false, a, false, b1, (short)0, d1, false, false);
    }

    const float bias0 = b_rel[m];
    const float bias1 = b_rel[m + 16];
#pragma unroll
    for (int r = 0; r < 8; ++r) {
        int row = base + r + 8 * half;
        if (row < nRows) {
            xout[(size_t)row * EMBED + m]      = lrelu(d0[r] + bias0);
            xout[(size_t)row * EMBED + m + 16] = lrelu(d1[r] + bias1);
        }
    }
}

// ---------------------------------------------------------------------------
// Batch MLP head: one thread per pair. Weights are wave-uniform -> s_loads.
// h = [x[home] , x[away]] (64) -> 8 -> 8 -> 8 -> 8 -> 3, lrelu each,
// then log_softmax over the 3 logits.
// ---------------------------------------------------------------------------
__global__ void mlp_head(const float* __restrict__ x,
                         const int* __restrict__ home,
                         const int* __restrict__ away,
                         const float* __restrict__ w0, const float* __restrict__ b0,
                         const float* __restrict__ w1, const float* __restrict__ b1,
                         const float* __restrict__ w2, const float* __restrict__ b2,
                         const float* __restrict__ w3, const float* __restrict__ b3,
                         const float* __restrict__ w4, const float* __restrict__ b4,
                         float* __restrict__ out, int nB) {
    int i = blockIdx.x * blockDim.x + threadIdx.x;
    if (i >= nB) return;

    float h[64];
    const float4* xh = (const float4*)(x + (size_t)home[i] * EMBED);
    const float4* xa = (const float4*)(x + (size_t)away[i] * EMBED);
#pragma unroll
    for (int q = 0; q < 8; ++q) {
        float4 v = xh[q];
        h[4 * q + 0] = v.x; h[4 * q + 1] = v.y; h[4 * q + 2] = v.z; h[4 * q + 3] = v.w;
    }
#pragma unroll
    for (int q = 0; q < 8; ++q) {
        float4 v = xa[q];
        h[32 + 4 * q + 0] = v.x; h[32 + 4 * q + 1] = v.y; h[32 + 4 * q + 2] = v.z; h[32 + 4 * q + 3] = v.w;
    }

    float t0[8];
#pragma unroll
    for (int o = 0; o < 8; ++o) {
        float acc = b0[o];
#pragma unroll
        for (int j = 0; j < 64; ++j) acc = fmaf(h[j], w0[o * 64 + j], acc);
        t0[o] = lrelu(acc);
    }
    float t1[8];
#pragma unroll
    for (int o = 0; o < 8; ++o) {
        float acc = b1[o];
#pragma unroll
        for (int j = 0; j < 8; ++j) acc = fmaf(t0[j], w1[o * 8 + j], acc);
        t1[o] = lrelu(acc);
    }
    float t2[8];
#pragma unroll
    for (int o = 0; o < 8; ++o) {
        float acc = b2[o];
#pragma unroll
        for (int j = 0; j < 8; ++j) acc = fmaf(t1[j], w2[o * 8 + j], acc);
        t2[o] = lrelu(acc);
    }
    float t3[8];
#pragma unroll
    for (int o = 0; o < 8; ++o) {
        float acc = b3[o];
#pragma unroll
        for (int j = 0; j < 8; ++j) acc = fmaf(t2[j], w3[o * 8 + j], acc);
        t3[o] = lrelu(acc);
    }
    float t4[3];
#pragma unroll
    for (int o = 0; o < 3; ++o) {
        float acc = b4[o];
#pragma unroll
        for (int j = 0; j < 8; ++j) acc = fmaf(t3[j], w4[o * 8 + j], acc);
        t4[o] = lrelu(acc);
    }

    float mx = fmaxf(t4[0], fmaxf(t4[1], t4[2]));
    float s  = expf(t4[0] - mx) + expf(t4[1] - mx) + expf(t4[2] - mx);
    float lz = logf(s) + mx;
    out[3 * (size_t)i + 0] = t4[0] - lz;
    out[3 * (size_t)i + 1] = t4[1] - lz;
    out[3 * (size_t)i + 2] = t4[2] - lz;
}

// ---------------------------------------------------------------------------
extern "C" void kernel_launch(void* const* d_in, const int* in_sizes, int n_in,
                              void* d_out, int out_size, void* d_ws, size_t ws_size,
                              hipStream_t stream) {
    const int*   src    = (const int*)d_in[0];
    const int*   dst    = src + NEDGES;
    const float* ew     = (const float*)d_in[1];
    const int*   home   = (const int*)d_in[2];
    const int*   away   = (const int*)d_in[3];
    const float* embed  = (const float*)d_in[4];
    const float* w_rel  = (const float*)d_in[5];
    const float* b_rel  = (const float*)d_in[6];
    const float* w_root = (const float*)d_in[7];
    const float* w0 = (const float*)d_in[8];  const float* b0 = (const float*)d_in[9];
    const float* w1 = (const float*)d_in[10]; const float* b1 = (const float*)d_in[11];
    const float* w2 = (const float*)d_in[12]; const float* b2 = (const float*)d_in[13];
    const float* w3 = (const float*)d_in[14]; const float* b3 = (const float*)d_in[15];
    const float* w4 = (const float*)d_in[16]; const float* b4 = (const float*)d_in[17];
    float* out = (float*)d_out;

    const size_t featElems = (size_t)NUM_TEAMS * EMBED;   // 3.2M floats
    float* x0  = (float*)d_ws;
    float* x1  = x0 + featElems;
    float* agg = x1 + featElems;

    const int n4       = (int)(featElems / 4);
    const int zeroBlks = (n4 + 255) / 256;

    const long long scatThreads = (long long)NEDGES * 32;
    const int scatBlks = (int)((scatThreads + 255) / 256);

    const int tiles    = (NUM_TEAMS + 15) / 16;           // 16-row WMMA tiles
    const int wavesPB  = 4;                               // 128 threads/block
    const int convBlks = (tiles + wavesPB - 1) / wavesPB;

    const float* cur = embed;
    float*       nxt = x0;
    for (int layer = 0; layer < NCONV; ++layer) {
        zero_f4<<<zeroBlks, 256, 0, stream>>>((float4*)agg, n4);
        edge_scatter<<<scatBlks, 256, 0, stream>>>(cur, src, dst, ew, agg, NEDGES);
        conv_transform<<<convBlks, wavesPB * 32, 0, stream>>>(
            cur, agg,
            w_rel + (size_t)layer * EMBED * EMBED,
            b_rel + (size_t)layer * EMBED,
            w_root + (size_t)layer * EMBED * EMBED,
            nxt, NUM_TEAMS);
        cur = nxt;
        nxt = (cur == x0) ? x1 : x0;
    }

    mlp_head<<<(BATCH + 255) / 256, 256, 0, stream>>>(
        cur, home, away, w0, b0, w1, b1, w2, b2, w3, b3, w4, b4, out, BATCH);
}